// Prototype_70454643524168
// MI455X (gfx1250) — compile-verified
//
#include <hip/hip_runtime.h>

#define N_ROWS 1000000
#define D_IN   128
#define D_OUT  64
#define NCLS   64

typedef __attribute__((ext_vector_type(2))) float v2f;
typedef __attribute__((ext_vector_type(8))) float v8f;

// Workspace layout (floats): [0 .. 8191] class sums R = S^T X  (64 x 128, row-major)
//                            [8192 .. 8255] per-class counts
#define WS_SUMS   0
#define WS_COUNTS (NCLS * D_IN)
#define WS_TOTAL  (NCLS * D_IN + NCLS)

__global__ void zero_ws_kernel(float* __restrict__ ws) {
    for (int i = blockIdx.x * blockDim.x + threadIdx.x; i < WS_TOTAL;
         i += gridDim.x * blockDim.x)
        ws[i] = 0.0f;
}

// Kernel 1: bandwidth-bound class accumulation of raw inputs.
// 256 threads = 8 wave32s; each wave processes one row per iteration.
// Lane l owns dims {l, l+32, l+64, l+96}: 4 coalesced 128B loads per wave,
// and ds_add_f32 addresses hit 32 distinct LDS banks per op (no conflicts).
__global__ void __launch_bounds__(256)
class_accum_kernel(const float* __restrict__ x,
                   const int*   __restrict__ labels,
                   float*       __restrict__ ws) {
    __shared__ float lsum[NCLS * D_IN];   // 32 KB
    __shared__ float lcnt[NCLS];

    const int tid = threadIdx.x;
    for (int i = tid; i < NCLS * D_IN; i += blockDim.x) lsum[i] = 0.0f;
    for (int i = tid; i < NCLS;        i += blockDim.x) lcnt[i] = 0.0f;
    __syncthreads();

    const int lane   = tid & 31;
    const int wave   = tid >> 5;
    const int wpb    = blockDim.x >> 5;
    const int gwave  = blockIdx.x * wpb + wave;
    const int nwaves = gridDim.x * wpb;

    for (int row = gwave; row < N_ROWS; row += nwaves) {
        const int lab = labels[row];                  // uniform across wave
        const float* __restrict__ r = x + (size_t)row * D_IN;
        const float v0 = r[lane];
        const float v1 = r[lane + 32];
        const float v2 = r[lane + 64];
        const float v3 = r[lane + 96];
        float* s = lsum + lab * D_IN;
        atomicAdd(&s[lane],      v0);   // ds_add_f32, conflict-free banks
        atomicAdd(&s[lane + 32], v1);
        atomicAdd(&s[lane + 64], v2);
        atomicAdd(&s[lane + 96], v3);
        if (lane == 0) atomicAdd(&lcnt[lab], 1.0f);
    }
    __syncthreads();

    // Flush block partials to global scratch with HW f32 atomics.
    for (int i = tid; i < NCLS * D_IN; i += blockDim.x)
        unsafeAtomicAdd(&ws[WS_SUMS + i], lsum[i]);
    for (int i = tid; i < NCLS; i += blockDim.x)
        unsafeAtomicAdd(&ws[WS_COUNTS + i], lcnt[i]);
}

// Kernel 2: protos = (R @ W + counts * b^T) / max(counts, 1)
// One block, 16 wave32s; wave w computes output tile (w/4, w%4) of the
// 64x64 result via V_WMMA_F32_16X16X4_F32 over K = 128 (32 steps).
//
// f32 16x4 A layout (ISA 7.12.2): lanes 0-15 hold M=0..15; VGPR0/1 carry
// K = {0,1} for lanes 0-15 and K = {2,3} for lanes 16-31.
// B (4x16) mirrors it: each VGPR holds one K-row striped across lanes 0-15,
// with lanes 16-31 holding the K+2 row.
__global__ void __launch_bounds__(512)
proto_wmma_kernel(const float* __restrict__ ws,
                  const float* __restrict__ W,     // [128, 64] row-major
                  const float* __restrict__ bias,  // [64]
                  float*       __restrict__ out) { // [64, 64] row-major
    const float* __restrict__ segsum = ws + WS_SUMS;
    const float* __restrict__ counts = ws + WS_COUNTS;

    const int tid  = threadIdx.x;
    const int lane = tid & 31;
    const int wave = tid >> 5;          // 0..15
    const int m0   = (wave >> 2) * 16;  // class-tile origin
    const int n0   = (wave & 3)  * 16;  // output-dim tile origin
    const int half = lane >> 4;         // 0: K pair {0,1}; 1: K pair {2,3}
    const int l    = lane & 15;

    v8f acc = {};
    #pragma unroll
    for (int k = 0; k < D_IN; k += 4) {
        const int ka = k + 2 * half;
        v2f a;
        a.x = segsum[(m0 + l) * D_IN + ka];      // A[m][ka]
        a.y = segsum[(m0 + l) * D_IN + ka + 1];  // A[m][ka+1]
        v2f bm;
        bm.x = W[ka       * D_OUT + n0 + l];     // B[ka][n]
        bm.y = W[(ka + 1) * D_OUT + n0 + l];     // B[ka+1][n]
        acc = __builtin_amdgcn_wmma_f32_16x16x4_f32(
            /*neg_a=*/false, a, /*neg_b=*/false, bm,
            /*c_mod=*/(short)0, acc, /*reuse_a=*/false, /*reuse_b=*/false);
    }

    const float bn = bias[n0 + l];
    #pragma unroll
    for (int r = 0; r < 8; ++r) {
        const int m     = m0 + r + 8 * half;     // C/D layout: VGPR r -> M=r (+8 for hi half)
        const float c   = counts[m];
        const float den = fmaxf(c, 1.0f);
        const float keep = (c > 0.0f) ? 1.0f : 0.0f;  // counts[m]/den
        out[m * D_OUT + n0 + l] = acc[r] / den + bn * keep;
    }
}

extern "C" void kernel_launch(void* const* d_in, const int* in_sizes, int n_in,
                              void* d_out, int out_size, void* d_ws, size_t ws_size,
                              hipStream_t stream) {
    const float* x      = (const float*)d_in[0];  // [N, 128]
    const int*   labels = (const int*)  d_in[1];  // [N]
    const float* W      = (const float*)d_in[2];  // [128, 64]
    const float* b      = (const float*)d_in[3];  // [64]
    float*       out    = (float*)d_out;          // [64, 64]
    float*       ws     = (float*)d_ws;

    (void)in_sizes; (void)n_in; (void)out_size; (void)ws_size;

    zero_ws_kernel<<<32, 256, 0, stream>>>(ws);
    class_accum_kernel<<<2048, 256, 0, stream>>>(x, labels, ws);
    proto_wmma_kernel<<<1, 512, 0, stream>>>(ws, W, b, out);
}